// SimpleGNN_81819126988817
// MI455X (gfx1250) — compile-verified
//
#include <hip/hip_runtime.h>

typedef __attribute__((ext_vector_type(2))) float v2f;
typedef __attribute__((ext_vector_type(8))) float v8f;

#define N_NODES 50000
#define N_EDGES 800000
#define IN_CH   128
#define HID_CH  128
#define OUT_CH  64

// ---------------- utility: zero a float buffer ----------------
__global__ __launch_bounds__(256) void zero_f32(float* __restrict__ p, int n) {
    int i = blockIdx.x * blockDim.x + threadIdx.x;
    if (i < n) p[i] = 0.0f;
}

// ---------------- degree count (deg[row[e]] += 1) ----------------
__global__ __launch_bounds__(256) void deg_count(const long long* __restrict__ row,
                                                 float* __restrict__ deg, int n_edges) {
    int e = blockIdx.x * blockDim.x + threadIdx.x;
    if (e < n_edges) atomicAdd(&deg[(int)row[e]], 1.0f);
}

__global__ __launch_bounds__(256) void deg_invert(float* __restrict__ deg, int n) {
    int i = blockIdx.x * blockDim.x + threadIdx.x;
    if (i < n) {
        float d = deg[i];
        deg[i] = (d > 0.0f) ? (1.0f / d) : 0.0f;
    }
}

// ---------------- WMMA f32 GEMM: D = rowscale(A) @ W^T + bias ----------------
// A: [M, K] row-major. W: [N, K] row-major (nn.Linear weight) -> B[k][n] = W[n][k].
// One wave computes a 16 x (16*NT) strip of D via V_WMMA_F32_16X16X4_F32,
// register-blocked along N: the A fragment is loaded once per k-step and reused
// across NT WMMAs (NT accumulators of 8 VGPRs each).
//
// A-fragment layout (32-bit A 16x4, ISA 7.12.2): lanes 0-15 -> M, lanes 16-31 -> M
// with K offset +2; VGPR0 = K even, VGPR1 = K odd. B-fragment mirrors with lanes
// mapping N. C/D: VGPR r -> M = r + 8*(lane>=16), lane%16 -> N.
template <int NT>
__global__ __launch_bounds__(32) void gemm_wmma_f32(
    const float* __restrict__ A, const float* __restrict__ W,
    const float* __restrict__ bias, const float* __restrict__ rowscale,
    float* __restrict__ D, int K, int N)
{
    const int lane  = threadIdx.x & 31;
    const int lrow  = lane & 15;     // M index within tile (A) / N index (B, D)
    const int khalf = lane >> 4;     // 0 or 1 -> +2 K offset

    const int m0 = blockIdx.x * 16;
    const int n0 = blockIdx.y * (16 * NT);

    const float ascale = (rowscale != nullptr) ? rowscale[m0 + lrow] : 1.0f;

    const float* __restrict__ arow = A + (size_t)(m0 + lrow) * K;
    const float* wrow[NT];
    #pragma unroll
    for (int t = 0; t < NT; ++t)
        wrow[t] = W + (size_t)(n0 + t * 16 + lrow) * K;

    v8f acc[NT];
    #pragma unroll
    for (int t = 0; t < NT; ++t) acc[t] = (v8f){};

    for (int k0 = 0; k0 < K; k0 += 4) {
        const int k = k0 + 2 * khalf;
        v2f a;
        a.x = arow[k + 0] * ascale;
        a.y = arow[k + 1] * ascale;
        #pragma unroll
        for (int t = 0; t < NT; ++t) {
            v2f b;
            b.x = wrow[t][k + 0];
            b.y = wrow[t][k + 1];
            // 8 args: (neg_a, A, neg_b, B, c_mod, C, reuse_a, reuse_b)
            acc[t] = __builtin_amdgcn_wmma_f32_16x16x4_f32(
                false, a, false, b, (short)0, acc[t], false, false);
        }
    }

    #pragma unroll
    for (int t = 0; t < NT; ++t) {
        const int n = n0 + t * 16 + lrow;
        const float bv = bias[n];
        #pragma unroll
        for (int r = 0; r < 8; ++r) {
            const int m = m0 + r + 8 * khalf;
            D[(size_t)m * N + n] = acc[t][r] + bv;
        }
    }
}

// ---------------- edge scatter: agg[col[e]] += h[row[e]] ----------------
// One wave per edge; lane handles 4 contiguous channels (float4 gather, 4 f32 atomics).
__global__ __launch_bounds__(256) void scatter_edges(
    const long long* __restrict__ row, const long long* __restrict__ col,
    const float* __restrict__ h, float* __restrict__ agg, int n_edges)
{
    const int e = blockIdx.x * (blockDim.x >> 5) + (threadIdx.x >> 5);
    if (e >= n_edges) return;
    const int lane = threadIdx.x & 31;

    const int src = (int)row[e];
    const int dst = (int)col[e];

    const float4 v = *(const float4*)(h + (size_t)src * HID_CH + lane * 4);
    float* __restrict__ d = agg + (size_t)dst * HID_CH + lane * 4;
    atomicAdd(d + 0, v.x);
    atomicAdd(d + 1, v.y);
    atomicAdd(d + 2, v.z);
    atomicAdd(d + 3, v.w);
}

extern "C" void kernel_launch(void* const* d_in, const int* in_sizes, int n_in,
                              void* d_out, int out_size, void* d_ws, size_t ws_size,
                              hipStream_t stream) {
    const float*     x    = (const float*)d_in[0];
    const long long* edge = (const long long*)d_in[1];   // [2, N_EDGES] int64
    const float*     W1   = (const float*)d_in[2];       // [HID, IN]
    const float*     b1   = (const float*)d_in[3];       // [HID]
    const float*     W2   = (const float*)d_in[4];       // [OUT, HID]
    const float*     b2   = (const float*)d_in[5];       // [OUT]
    float*           out  = (float*)d_out;               // [N, OUT]

    // Workspace layout: h | agg | deg  (contiguous f32)
    float* h   = (float*)d_ws;
    float* agg = h   + (size_t)N_NODES * HID_CH;
    float* deg = agg + (size_t)N_NODES * HID_CH;

    const long long* erow = edge;
    const long long* ecol = edge + N_EDGES;

    // 1) zero agg + deg (contiguous region)
    {
        int nz = N_NODES * HID_CH + N_NODES;
        zero_f32<<<(nz + 255) / 256, 256, 0, stream>>>(agg, nz);
    }

    // 2) degree count from ROW, then invert in place
    deg_count<<<(N_EDGES + 255) / 256, 256, 0, stream>>>(erow, deg, N_EDGES);
    deg_invert<<<(N_NODES + 255) / 256, 256, 0, stream>>>(deg, N_NODES);

    // 3) h = x @ W1^T + b1  (WMMA f32, 16x64 strip per wave)
    {
        dim3 grid(N_NODES / 16, HID_CH / 64);   // 3125 x 2
        gemm_wmma_f32<4><<<grid, 32, 0, stream>>>(x, W1, b1, nullptr, h, IN_CH, HID_CH);
    }

    // 4) agg[dst] += h[src]  (one wave per edge)
    {
        int waves_per_block = 256 / 32;
        int blocks = (N_EDGES + waves_per_block - 1) / waves_per_block;
        scatter_edges<<<blocks, 256, 0, stream>>>(erow, ecol, h, agg, N_EDGES);
    }

    // 5) out = (deg_inv * agg) @ W2^T + b2  (normalization fused into A load)
    {
        dim3 grid(N_NODES / 16, OUT_CH / 64);   // 3125 x 1
        gemm_wmma_f32<4><<<grid, 32, 0, stream>>>(agg, W2, b2, deg, out, HID_CH, OUT_CH);
    }
}